// Quantumnet_12378095747836
// MI455X (gfx1250) — compile-verified
//
#include <hip/hip_runtime.h>
#include <hip/hip_bf16.h>
#include <math.h>

typedef __attribute__((ext_vector_type(2))) float v2f;
typedef __attribute__((ext_vector_type(8))) float v8f;

#define BATCH 16384
#define DIM   3072
#define NQ    4
#define NC    3
#define TILE_M 16
#define KSPLIT 2                      // waves cooperating on one M-tile (K halves)
#define KCHUNK (DIM / KSPLIT)         // 1536
#define WAVES_PER_BLOCK 4
#define TILES_PER_BLOCK (WAVES_PER_BLOCK / KSPLIT)   // 2
#define THREADS (WAVES_PER_BLOCK * 32)
#define PF_ELEMS 1024                 // prefetch distance: 4 KB ahead per row

__global__ __launch_bounds__(THREADS) void qnet_fused(
    const float* __restrict__ x, const float* __restrict__ pre_w,
    const float* __restrict__ pre_b, const float* __restrict__ q_params,
    const float* __restrict__ post_w, const float* __restrict__ post_b,
    float* __restrict__ out)
{
    __shared__ float hh[WAVES_PER_BLOCK][TILE_M][NQ];

    const int lane = threadIdx.x & 31;
    const int wave = threadIdx.x >> 5;
    const int kw   = wave & (KSPLIT - 1);   // which K half this wave reduces
    const int tloc = wave >> 1;             // tile within block
    const int tile = blockIdx.x * TILES_PER_BLOCK + tloc;
    const int row0 = tile * TILE_M;

    const int n    = lane & 15;   // A-row within tile / B-column index
    const int hi   = lane >> 4;   // lanes 16-31 hold K pair {2,3} of each step
    const int koff = hi * 2;
    const int kbase = kw * KCHUNK;

    // A: x[row0+n, kbase + k + koff .. +1]   (16x4 f32 A-matrix layout)
    const float* aptr = x + (size_t)(row0 + n) * DIM + kbase + koff;
    // B: pre_w[n, ...]; columns n>=4 alias row n&3 -> garbage lands only in
    // D columns >=4 which are never read.
    const float* bptr = pre_w + (size_t)(n & 3) * DIM + kbase + koff;

    v8f acc = {};
    #pragma unroll 16
    for (int k = 0; k < KCHUNK; k += 4) {
        if ((k & 63) == 0) {
            // pull the next 2 cachelines of this row toward the WGP ahead of
            // the demand loads (global_prefetch_b8, speculative, no LOADcnt)
            __builtin_prefetch(aptr + k + PF_ELEMS);
            __builtin_prefetch(aptr + k + PF_ELEMS + 32);
        }
        v2f a = __builtin_nontemporal_load((const v2f*)(aptr + k)); // streamed once
        v2f b = *(const v2f*)(bptr + k);                            // 48KB, cache-hot
        acc = __builtin_amdgcn_wmma_f32_16x16x4_f32(
            false, a, false, b, (short)0, acc, false, false);
    }

    // D layout: VGPR r holds D[m = r + 8*hi][n = lane&15]. Keep cols 0..3.
    if (n < NQ) {
        #pragma unroll
        for (int r = 0; r < 8; ++r)
            hh[wave][r + 8 * hi][n] = acc[r];
    }
    __syncthreads();

    // one wave of each K-pair runs the epilogue; lanes 0..15 = rows
    if (kw == 0 && lane < TILE_M) {
        const int row = row0 + lane;
        float h[NQ];
        #pragma unroll
        for (int w = 0; w < NQ; ++w) {
            float d = hh[wave][lane][w] + hh[wave + 1][lane][w]; // combine K halves
            h[w] = tanhf(d + pre_b[w]) * 1.0471975511965976f;    // * pi/3
        }

        // ---- 4-qubit statevector sim, fully register-resident ----
        float sr[16], si[16];
        #pragma unroll
        for (int i = 0; i < 16; ++i) { sr[i] = 0.0f; si[i] = 0.0f; }
        sr[0] = 1.0f;

        auto ry = [&](int stride, float c, float s) {
            #pragma unroll
            for (int i = 0; i < 16; ++i) {
                if ((i & stride) == 0) {
                    const int j = i | stride;
                    float ar = sr[i], ai = si[i], br = sr[j], bi = si[j];
                    sr[i] = c * ar - s * br;  si[i] = c * ai - s * bi;
                    sr[j] = s * ar + c * br;  si[j] = s * ai + c * bi;
                }
            }
        };
        auto cnot = [&](int sc, int st) {
            #pragma unroll
            for (int i = 0; i < 16; ++i) {
                if ((i & sc) && !(i & st)) {
                    const int j = i | st;
                    float tr = sr[i], ti = si[i];
                    sr[i] = sr[j]; si[i] = si[j];
                    sr[j] = tr;    si[j] = ti;
                }
            }
        };

        // encoding RY(h[w]) on wire w (wire w -> bit stride 8>>w)
        #pragma unroll
        for (int w = 0; w < NQ; ++w) {
            float s, c;
            sincosf(0.5f * h[w], &s, &c);
            ry(8 >> w, c, s);
        }
        // 3 layers of [RY(q_params[8L+w]) w=0..3 ; CNOT(w,(w+1)%4) w=0..3]
        #pragma unroll
        for (int L = 0; L < 3; ++L) {
            #pragma unroll
            for (int w = 0; w < NQ; ++w) {
                float s, c;
                sincosf(0.5f * q_params[8 * L + w], &s, &c);
                ry(8 >> w, c, s);
            }
            #pragma unroll
            for (int w = 0; w < NQ; ++w)
                cnot(8 >> w, 8 >> ((w + 1) & 3));
        }

        // <Z_w> expectation values
        float z[NQ] = {0.0f, 0.0f, 0.0f, 0.0f};
        #pragma unroll
        for (int i = 0; i < 16; ++i) {
            float p = sr[i] * sr[i] + si[i] * si[i];
            #pragma unroll
            for (int w = 0; w < NQ; ++w)
                z[w] += (i & (8 >> w)) ? -p : p;
        }

        // tiny output head: [NQ] @ post_w^T + post_b
        #pragma unroll
        for (int c = 0; c < NC; ++c) {
            float o = post_b[c];
            #pragma unroll
            for (int w = 0; w < NQ; ++w)
                o += z[w] * post_w[c * NQ + w];
            out[(size_t)row * NC + c] = o;
        }
    }
}

extern "C" void kernel_launch(void* const* d_in, const int* in_sizes, int n_in,
                              void* d_out, int out_size, void* d_ws, size_t ws_size,
                              hipStream_t stream) {
    (void)in_sizes; (void)n_in; (void)out_size; (void)d_ws; (void)ws_size;
    const float* x        = (const float*)d_in[0];
    const float* pre_w    = (const float*)d_in[1];
    const float* pre_b    = (const float*)d_in[2];
    const float* q_params = (const float*)d_in[3];
    const float* post_w   = (const float*)d_in[4];
    const float* post_b   = (const float*)d_in[5];
    float* out = (float*)d_out;

    const int tiles  = BATCH / TILE_M;             // 1024 tiles
    const int blocks = tiles / TILES_PER_BLOCK;    // 512 blocks x 128 threads
    qnet_fused<<<blocks, THREADS, 0, stream>>>(x, pre_w, pre_b, q_params,
                                               post_w, post_b, out);
}